// ForwardSumLossWithBlank_51161650430343
// MI455X (gfx1250) — compile-verified
//
#include <hip/hip_runtime.h>
#include <cstdint>

#define DEV __device__ __forceinline__

static constexpr float NEGV = -1000000000.0f;
static constexpr int TT   = 2000;   // mel frames (fixed by harness)
static constexpr int CC   = 400;    // text classes (fixed by harness)
static constexpr int SBLK = 832;    // 26 wave32s, covers S = 2*400+1 = 801 states
static constexpr int PF   = 6;      // async prefetch depth (rows in flight)
static constexpr int NBUF = PF + 1; // LDS row ring buffers

// ---- CDNA5 async global->LDS path (builtin confirmed present by round-1 diag) ----
#if defined(__HIP_DEVICE_COMPILE__) && __has_builtin(__builtin_amdgcn_global_load_async_to_lds_b128)
#define ASYNC_MODE 1   // clang builtin
#elif defined(__HIP_DEVICE_COMPILE__)
#define ASYNC_MODE 2   // inline asm
#else
#define ASYNC_MODE 0   // host parse / fallback sync copy
#endif

typedef int v4i_ __attribute__((vector_size(16)));
typedef __attribute__((address_space(1))) v4i_* as1_v4i_p;
typedef __attribute__((address_space(3))) v4i_* as3_v4i_p;

DEV void issue_row(const float* __restrict__ g, float* l, int tid) {
  if (tid < CC / 4) {                       // 100 lanes x 16B = 400 floats
#if ASYNC_MODE == 1
    __builtin_amdgcn_global_load_async_to_lds_b128(
        (as1_v4i_p)(g + tid * 4),           // global src (addrspace(1), const dropped)
        (as3_v4i_p)(l + tid * 4),           // LDS dst (addrspace(3))
        0, 0);
#elif ASYNC_MODE == 2
    unsigned lofs = (unsigned)(uintptr_t)(__attribute__((address_space(3))) void*)(l + tid * 4);
    const void* gaddr = (const void*)(g + tid * 4);
    asm volatile("global_load_async_to_lds_b128 %0, %1, off"
                 :: "v"(lofs), "v"(gaddr) : "memory");
#else
    float4 v = *(const float4*)(g + tid * 4);
    *(float4*)(l + tid * 4) = v;
#endif
  }
}

template <int N> DEV void wait_rows() {
#if ASYNC_MODE != 0
#if __has_builtin(__builtin_amdgcn_s_wait_asynccnt)
  __builtin_amdgcn_s_wait_asynccnt(N);
#else
  asm volatile("s_wait_asynccnt %0" :: "n"(N) : "memory");
#endif
#endif
}

// ---------------- Kernel 1: masked log-sum-exp per (b,t) row ----------------
// One wave32 per row; 8 rows per 256-thread block. Streams 102 MB once; data
// lands in the 192 MB L2 so kernel 2's re-read is L2-resident.
__global__ __launch_bounds__(256) void lse_kernel(const float* __restrict__ logits,
                                                  const int* __restrict__ tlen,
                                                  float* __restrict__ lse,
                                                  int nrows) {
  const int wid  = threadIdx.x >> 5;
  const int lane = threadIdx.x & 31;
  const long long row = (long long)blockIdx.x * 8 + wid;
  if (row >= nrows) return;
  const int b = (int)(row / TT);
  const int L = tlen[b];
  const float* __restrict__ x = logits + row * CC;

  float v[13];
  float mx = NEGV;
#pragma unroll
  for (int k = 0; k < 13; ++k) {
    const int c = lane + k * 32;
    float xv = (c < CC) ? x[c] : NEGV;
    v[k] = (c < L) ? xv : NEGV;           // mask classes >= text_len
    mx = fmaxf(mx, v[k]);
  }
#pragma unroll
  for (int off = 16; off; off >>= 1) mx = fmaxf(mx, __shfl_xor(mx, off, 32));
  float sum = 0.0f;
#pragma unroll
  for (int k = 0; k < 13; ++k) sum += __expf(v[k] - mx);
#pragma unroll
  for (int off = 16; off; off >>= 1) sum += __shfl_xor(sum, off, 32);
  if (lane == 0) lse[row] = mx + __logf(sum);
}

// ---------------- Kernel 2: sequential CTC alpha scan, 1 block / sample -----
// Thread s owns alpha state s. Ping-pong alpha (1 barrier/step). Row t+PF is
// async-DMA'd into an LDS ring while step t computes; s_wait_asynccnt(PF-1)
// before the barrier guarantees row t+1 for every wave at step t+1.
__global__ __launch_bounds__(SBLK) void ctc_scan_kernel(const float* __restrict__ logits,
                                                        const float* __restrict__ lse,
                                                        const int* __restrict__ tlen,
                                                        const int* __restrict__ mlen,
                                                        float* __restrict__ losses) {
  __shared__ float alphaS[2][SBLK];
  __shared__ float lseS[TT];
  __shared__ __align__(16) float rowS[NBUF][CC];

  const int b = blockIdx.x;
  const int s = threadIdx.x;
  const int L = tlen[b];
  const int M = mlen[b];                       // >= 1024 per harness
  const float* __restrict__ base = logits + (size_t)b * TT * CC;
  const float* __restrict__ lseb = lse + (size_t)b * TT;

  for (int t = s; t < M; t += SBLK) lseS[t] = lseb[t];

  // Prime pipeline: rows 0..PF in flight; complete rows 0 and 1.
#pragma unroll
  for (int r = 0; r <= PF; ++r) issue_row(base + (size_t)r * CC, rowS[r % NBUF], s);
  wait_rows<PF - 1>();
  __syncthreads();

  const int  lab  = (s & 1) ? ((s - 1) >> 1) : 0;   // blank=0 on even states
  const bool skip = (s & 1) && (s >= 3);

  int pp = 0;
  {
    const float lp0 = rowS[0][0] - lseS[0];          // ext labels 0,1 both -> col 0
    alphaS[pp][s] = (s <= 1) ? lp0 : NEGV;
  }
  __syncthreads();

  for (int t = 1; t < M; ++t) {
    const float a0 = alphaS[pp][s];
    const float a1 = (s >= 1) ? alphaS[pp][s - 1] : NEGV;
    const float a2 = skip ? alphaS[pp][s - 2] : NEGV;

    const int  r    = t + PF;
    const bool more = (r < M);
    if (more) issue_row(base + (size_t)r * CC, rowS[r % NBUF], s);

    const float x  = (lab < L) ? rowS[t % NBUF][lab] : NEGV;
    const float lp = x - lseS[t];
    const float m  = fmaxf(fmaxf(a0, a1), a2);
    const float nv = m + __logf(__expf(a0 - m) + __expf(a1 - m) + __expf(a2 - m)) + lp;
    alphaS[pp ^ 1][s] = nv;

    if (more) wait_rows<PF - 1>(); else wait_rows<0>();
    __syncthreads();
    pp ^= 1;
  }

  if (s == 0) {
    const int   se = 2 * L;
    const float aN = alphaS[pp][se];
    const float aM = alphaS[pp][se - 1];
    const float mm = fmaxf(aN, aM);
    const float ll = mm + __logf(__expf(aN - mm) + __expf(aM - mm));
    losses[b] = (ll > 0.5f * NEGV) ? (-ll / (float)L) : 0.0f;
  }
}

// ---------------- Kernel 3: deterministic mean over batch -------------------
__global__ __launch_bounds__(32) void mean_kernel(const float* __restrict__ losses,
                                                  float* __restrict__ out, int B) {
  float v = (threadIdx.x < B) ? losses[threadIdx.x] : 0.0f;
#pragma unroll
  for (int off = 16; off; off >>= 1) v += __shfl_xor(v, off, 32);
  if (threadIdx.x == 0) out[0] = v / (float)B;
}

extern "C" void kernel_launch(void* const* d_in, const int* in_sizes, int n_in,
                              void* d_out, int out_size, void* d_ws, size_t ws_size,
                              hipStream_t stream) {
  (void)n_in; (void)out_size; (void)ws_size;
  const float* attn = (const float*)d_in[0];   // [B,1,T,C] fp32
  const int* tlen   = (const int*)d_in[1];     // [B] int32
  const int* mlen   = (const int*)d_in[2];     // [B] int32
  const int B = in_sizes[1];
  const int nrows = B * TT;

  float* lse    = (float*)d_ws;                // B*T floats (256 KB for B=32)
  float* losses = lse + (size_t)B * TT;        // B floats

  lse_kernel<<<(nrows + 7) / 8, 256, 0, stream>>>(attn, tlen, lse, nrows);
  ctc_scan_kernel<<<B, SBLK, 0, stream>>>(attn, lse, tlen, mlen, losses);
  mean_kernel<<<1, 32, 0, stream>>>(losses, (float*)d_out, B);
}